// SCRFDTDMMPostModel_82806969467112
// MI455X (gfx1250) — compile-verified
//
#include <hip/hip_runtime.h>
#include <math.h>

typedef __attribute__((ext_vector_type(2))) float v2f;
typedef __attribute__((ext_vector_type(8))) float v8f;

#define BB 32
#define AA 3200
#define NOBJ 15
#define NPAR 59
#define NK 50          // 40 shp + 10 exp coefficients
#define KPAD 52        // K padded to multiple of 4
#define NV 204         // vertex components (68*3)
#define NPADV 208      // N padded to multiple of 16
#define KP_THRES 0.5f
#define IOU_THRES 0.45f
#define INVRESIZE (1.0f / 640.0f)

// ---------------- workspace layout (floats) ----------------
// [0,480)        topv  (B*15)
// [480,960)      topi  (B*15, int)
// [960,1168)     u_pad (208)
// [1168,11984)   baseT (52*208), baseT[k*208+n] = base[n][k]

__global__ void scrfd_prep_kernel(const float* __restrict__ u_base,
                                  const float* __restrict__ shp_base,
                                  const float* __restrict__ exp_base,
                                  float* __restrict__ wsf) {
    int i = blockIdx.x * blockDim.x + threadIdx.x;
    float* upad  = wsf + 960;
    float* baseT = wsf + 1168;
    if (i < KPAD * NPADV) {
        int k = i / NPADV, n = i % NPADV;
        float v = 0.0f;
        if (n < NV) {
            if (k < 40)       v = shp_base[n * 40 + k];
            else if (k < NK)  v = exp_base[n * 10 + (k - 40)];
        }
        baseT[i] = v;
    } else if (i < KPAD * NPADV + NPADV) {
        int n = i - KPAD * NPADV;
        upad[n] = (n < NV) ? u_base[n] : 0.0f;
    }
}

// ---------------- per-image top-15 selection ----------------
__global__ void __launch_bounds__(256)
scrfd_topk_kernel(const float* __restrict__ cls_preds, float* __restrict__ wsf) {
    __shared__ float sc[AA * 2];
    __shared__ float rv[256];
    __shared__ int   ri[256];
    const int b = blockIdx.x, tid = threadIdx.x;
    const float* cls = cls_preds + (size_t)b * AA * 2;
    for (int i = tid; i < AA * 2; i += 256) {
        float x = cls[i];
        sc[i] = 1.0f / (1.0f + expf(-x));      // sigmoid
    }
    __syncthreads();
    float* topv = wsf;
    int*   topi = (int*)(wsf + 480);
    for (int r = 0; r < NOBJ; ++r) {
        float bv = -INFINITY; int bi = 0x7fffffff;
        for (int i = tid; i < AA * 2; i += 256) {
            float v = sc[i];
            if (v > bv || (v == bv && i < bi)) { bv = v; bi = i; }
        }
        rv[tid] = bv; ri[tid] = bi;
        __syncthreads();
        for (int s = 128; s > 0; s >>= 1) {
            if (tid < s) {
                float ov = rv[tid + s]; int oi = ri[tid + s];
                if (ov > rv[tid] || (ov == rv[tid] && oi < ri[tid])) {
                    rv[tid] = ov; ri[tid] = oi;
                }
            }
            __syncthreads();
        }
        if (tid == 0) {
            topv[b * NOBJ + r] = rv[0];
            topi[b * NOBJ + r] = ri[0];
            sc[ri[0]] = -INFINITY;             // exclude for next round
        }
        __syncthreads();
    }
}

// ---------------- main: decode + WMMA verts + landmarks + NMS ----------------
__global__ void __launch_bounds__(416)
scrfd_main_kernel(const float* __restrict__ bbox_preds,
                  const float* __restrict__ param_preds,
                  const float* __restrict__ param_kps,
                  const float* __restrict__ origin_shapes,
                  const float* __restrict__ pms,
                  const float* __restrict__ wsf,
                  float* __restrict__ out) {
    __shared__ float cLDS[16 * KPAD];       // A tile: 16 anchors x 52 coeffs
    __shared__ float vertsLDS[16 * NPADV];  // C tile: 16 anchors x 208 verts
    __shared__ float lnmkLDS[NOBJ * 136];   // 15 x 68 x 2 landmarks
    __shared__ float RLDS[NOBJ * 9];
    __shared__ float boxLDS[NOBJ * 4];
    __shared__ float kpsLDS[NOBJ * 2];
    __shared__ float topvLDS[NOBJ];
    __shared__ int   keepLDS[NOBJ];

    const int b = blockIdx.x, tid = threadIdx.x;
    const float* topv  = wsf;
    const int*   topi  = (const int*)(wsf + 480);
    const float* upad  = wsf + 960;
    const float* baseT = wsf + 1168;

    // zero the A tile (covers K-pad cols and pad row 15)
    for (int i = tid; i < 16 * KPAD; i += 416) cLDS[i] = 0.0f;
    __syncthreads();

    // decode 15 selected anchors
    if (tid < NOBJ) {
        int sel   = topi[b * NOBJ + tid];
        int a_idx = sel >> 1;
        int cell  = a_idx >> 1;               // NUM_ANCHORS = 2
        float acx = (float)(cell % 40) * 16.0f;
        float acy = (float)(cell / 40) * 16.0f;
        float r0 = origin_shapes[b * 2 + 0] * INVRESIZE;
        float r1 = origin_shapes[b * 2 + 1] * INVRESIZE;
        const float* bb = bbox_preds + ((size_t)b * AA + a_idx) * 4;
        float d0 = bb[0] * 16.0f, d1 = bb[1] * 16.0f;
        float d2 = bb[2] * 16.0f, d3 = bb[3] * 16.0f;
        boxLDS[tid * 4 + 0] = (acy - d1) * r0;   // y1
        boxLDS[tid * 4 + 1] = (acx - d0) * r1;   // x1
        boxLDS[tid * 4 + 2] = (acy + d3) * r0;   // y2
        boxLDS[tid * 4 + 3] = (acx + d2) * r1;   // x2
        const float* pk = param_kps + ((size_t)b * AA + a_idx) * 2;
        float kx = pk[0] * 16.0f + acx, ky = pk[1] * 16.0f + acy;
        kpsLDS[tid * 2 + 0] = (ky + 0.5f) * r0;
        kpsLDS[tid * 2 + 1] = (kx + 0.5f) * r1;
        const float* pp = param_preds + ((size_t)b * AA + a_idx) * NPAR;
        for (int p = 0; p < 9; ++p)
            RLDS[tid * 9 + p] = pp[p] * pms[NPAR + p] + pms[p];
        for (int k = 0; k < NK; ++k)
            cLDS[tid * KPAD + k] = pp[9 + k] * pms[NPAR + 9 + k] + pms[9 + k];
        topvLDS[tid] = topv[b * NOBJ + tid];
    }
    __syncthreads();

    // ---- WMMA F32 16x16x4: verts[16 anchors][16 cols] per wave ----
    const int lane  = tid & 31;
    const int wave  = tid >> 5;            // 0..12 -> N-tile
    const int n0    = wave * 16;
    const int laneM = lane & 15;
    const int kh2   = (lane >> 4) << 1;    // 0 or 2
    v8f acc = {};
#pragma unroll
    for (int s = 0; s < 13; ++s) {
        int kb = 4 * s + kh2;
        v2f a, bm;
        a.x  = cLDS[laneM * KPAD + kb];
        a.y  = cLDS[laneM * KPAD + kb + 1];
        bm.x = baseT[(kb    ) * NPADV + n0 + laneM];
        bm.y = baseT[(kb + 1) * NPADV + n0 + laneM];
        acc = __builtin_amdgcn_wmma_f32_16x16x4_f32(
            false, a, false, bm, (short)0, acc, false, false);
    }
    {
        float uval = upad[n0 + laneM];
        int mhi = (lane >> 4) << 3;        // 0 or 8
#pragma unroll
        for (int j = 0; j < 8; ++j)
            vertsLDS[(j + mhi) * NPADV + n0 + laneM] = acc[j] + uval;
    }
    __syncthreads();

    // ---- landmarks: lnmks[j] = (verts_j . R[1], verts_j . R[0]) ----
    for (int t = tid; t < NOBJ * 68; t += 416) {
        int m = t / 68, j = t % 68;
        float vx = vertsLDS[m * NPADV + j * 3 + 0];
        float vy = vertsLDS[m * NPADV + j * 3 + 1];
        float vz = vertsLDS[m * NPADV + j * 3 + 2];
        const float* R = &RLDS[m * 9];
        lnmkLDS[m * 136 + j * 2 + 0] = vx * R[3] + vy * R[4] + vz * R[5]; // e=1
        lnmkLDS[m * 136 + j * 2 + 1] = vx * R[0] + vy * R[1] + vz * R[2]; // e=0
    }
    __syncthreads();

    // ---- per-anchor scale + shift (threads 0..14); NMS (thread 15) ----
    if (tid < NOBJ) {
        int m = tid;
        float tl0 = INFINITY, tl1 = INFINITY, br0 = -INFINITY, br1 = -INFINITY;
        for (int j = 0; j < 68; ++j) {
            float l0 = lnmkLDS[m * 136 + j * 2 + 0];
            float l1 = lnmkLDS[m * 136 + j * 2 + 1];
            tl0 = fminf(tl0, l0); br0 = fmaxf(br0, l0);
            tl1 = fminf(tl1, l1); br1 = fmaxf(br1, l1);
        }
        float wh0 = boxLDS[m * 4 + 2] - boxLDS[m * 4 + 0];
        float wh1 = boxLDS[m * 4 + 3] - boxLDS[m * 4 + 1];
        float s0 = wh0 / (br0 - tl0 + 1e-9f);
        float s1 = wh1 / (br1 - tl1 + 1e-9f);
        float k0 = kpsLDS[m * 2 + 0], k1 = kpsLDS[m * 2 + 1];
        for (int j = 0; j < 68; ++j) {
            lnmkLDS[m * 136 + j * 2 + 0] = lnmkLDS[m * 136 + j * 2 + 0] * s0 + k0;
            lnmkLDS[m * 136 + j * 2 + 1] = lnmkLDS[m * 136 + j * 2 + 1] * s1 + k1;
        }
    } else if (tid == NOBJ) {
        bool keep[NOBJ];
        float y1[NOBJ], x1[NOBJ], y2[NOBJ], x2[NOBJ], ar[NOBJ];
        for (int i = 0; i < NOBJ; ++i) {
            y1[i] = boxLDS[i * 4 + 0]; x1[i] = boxLDS[i * 4 + 1];
            y2[i] = boxLDS[i * 4 + 2]; x2[i] = boxLDS[i * 4 + 3];
            ar[i] = (y2[i] - y1[i]) * (x2[i] - x1[i]);
            keep[i] = topvLDS[i] > KP_THRES;
        }
        for (int i = 0; i < NOBJ; ++i) {
            bool sup = false;
            for (int j = 0; j < i; ++j) {
                if (!keep[j]) continue;
                float iy = fminf(y2[j], y2[i]) - fmaxf(y1[j], y1[i]);
                float ix = fminf(x2[j], x2[i]) - fmaxf(x1[j], x1[i]);
                iy = fmaxf(0.0f, iy); ix = fmaxf(0.0f, ix);
                float inter = iy * ix;
                float iou = inter / (ar[j] + ar[i] - inter + 1e-9f);
                if (iou > IOU_THRES) sup = true;
            }
            keep[i] = keep[i] && !sup;
        }
        for (int i = 0; i < NOBJ; ++i) keepLDS[i] = keep[i] ? 1 : 0;
    }
    __syncthreads();

    // ---- write outputs: boxes[B,15,4] | scores[B,15] | lnmks[B,15,68,2] ----
    const float INF = INFINITY;
    for (int t = tid; t < NOBJ * 4; t += 416) {
        int m = t / 4;
        out[b * (NOBJ * 4) + t] = keepLDS[m] ? boxLDS[t] : INF;
    }
    for (int t = tid; t < NOBJ; t += 416) {
        out[BB * NOBJ * 4 + b * NOBJ + t] = keepLDS[t] ? topvLDS[t] : INF;
    }
    for (int t = tid; t < NOBJ * 136; t += 416) {
        int m = t / 136;
        out[BB * NOBJ * 5 + b * (NOBJ * 136) + t] = keepLDS[m] ? lnmkLDS[t] : INF;
    }
}

extern "C" void kernel_launch(void* const* d_in, const int* in_sizes, int n_in,
                              void* d_out, int out_size, void* d_ws, size_t ws_size,
                              hipStream_t stream) {
    const float* cls_preds     = (const float*)d_in[0];
    const float* bbox_preds    = (const float*)d_in[1];
    const float* param_preds   = (const float*)d_in[2];
    const float* param_kps     = (const float*)d_in[3];
    const float* origin_shapes = (const float*)d_in[4];
    const float* pms           = (const float*)d_in[5];
    const float* u_base        = (const float*)d_in[6];
    const float* shp_base      = (const float*)d_in[7];
    const float* exp_base      = (const float*)d_in[8];
    float* out = (float*)d_out;
    float* wsf = (float*)d_ws;

    int prep_elems = KPAD * NPADV + NPADV;
    scrfd_prep_kernel<<<(prep_elems + 255) / 256, 256, 0, stream>>>(
        u_base, shp_base, exp_base, wsf);
    scrfd_topk_kernel<<<BB, 256, 0, stream>>>(cls_preds, wsf);
    scrfd_main_kernel<<<BB, 416, 0, stream>>>(
        bbox_preds, param_preds, param_kps, origin_shapes, pms, wsf, out);
}